// SelfAttention_1580547972937
// MI455X (gfx1250) — compile-verified
//
#include <hip/hip_runtime.h>
#include <hip/hip_bf16.h>
#include <math.h>

// ---------------------------------------------------------------------------
// Problem: B=256, D=1024 fp32 "scalar-energy" self attention.
//   q,k,v = x @ W{q,k,v}.T + b{q,k,v}
//   out[b,j] = gamma * softmax_i(q[b,j]*k[b,i]) . v[b,:] + x[b,j]
// Key identity: the softmax row depends on the scalar s = q[b,j] only, so
//   rowmax = max(s*kmax_b, s*kmin_b)
// and the B x D x D energy tensor is never materialized (saves ~3 GB traffic).
// ---------------------------------------------------------------------------

typedef float v2f __attribute__((ext_vector_type(2)));
typedef float v8f __attribute__((ext_vector_type(8)));

#define DD 1024
#define BB 256

// ---------------------------------------------------------------------------
// Kernel 1: O = X @ W^T + bias  using V_WMMA_F32_16X16X4_F32 (exact fp32).
// One wave (32 threads) computes a 16x64 strip: 4 C tiles reuse one A frag.
//
// ISA VGPR layouts (CDNA5 ISA):
//  A 16x4 (MxK):  lane = M + 16*(K/2), vgpr = K%2        -> float2 per lane
//  B 4x16 (KxN):  vgpr v holds K = v + 2*(lane/16), N = lane%16 (mirrors C/D
//                 "row r and r+half share a VGPR" convention) -> float2 per lane
//  C/D 16x16:     vgpr r holds M = r + 8*(lane/16), N = lane%16
// ---------------------------------------------------------------------------
__global__ __launch_bounds__(32)
void qkv_wmma_f32(const float* __restrict__ X,
                  const float* __restrict__ Wq, const float* __restrict__ bq,
                  const float* __restrict__ Wk, const float* __restrict__ bk,
                  const float* __restrict__ Wv, const float* __restrict__ bv,
                  float* __restrict__ Qo, float* __restrict__ Ko,
                  float* __restrict__ Vo)
{
    const int z = blockIdx.z;
    const float* __restrict__ W    = (z == 0) ? Wq : (z == 1) ? Wk : Wv;
    const float* __restrict__ bias = (z == 0) ? bq : (z == 1) ? bk : bv;
    float* __restrict__ O          = (z == 0) ? Qo : (z == 1) ? Ko : Vo;

    const int lane = threadIdx.x;          // 0..31
    const int row  = lane & 15;            // M (for A) / N (for B) within tile
    const int koff = (lane >> 4) << 1;     // lanes 16..31 carry K+2

    const int m0 = blockIdx.x * 16;        // 256/16 = 16 tiles
    const int n0 = blockIdx.y * 64;        // 1024/64 = 16 strips

    const float* __restrict__ ap  = X + (size_t)(m0 + row) * DD + koff;
    const float* __restrict__ bp0 = W + (size_t)(n0 +  0 + row) * DD + koff;
    const float* __restrict__ bp1 = W + (size_t)(n0 + 16 + row) * DD + koff;
    const float* __restrict__ bp2 = W + (size_t)(n0 + 32 + row) * DD + koff;
    const float* __restrict__ bp3 = W + (size_t)(n0 + 48 + row) * DD + koff;

    v8f c0 = {}, c1 = {}, c2 = {}, c3 = {};

    for (int kb = 0; kb < DD; kb += 4) {
        v2f a  = *(const v2f*)(ap  + kb);
        v2f b0 = *(const v2f*)(bp0 + kb);
        v2f b1 = *(const v2f*)(bp1 + kb);
        v2f b2 = *(const v2f*)(bp2 + kb);
        v2f b3 = *(const v2f*)(bp3 + kb);
        c0 = __builtin_amdgcn_wmma_f32_16x16x4_f32(false, a, false, b0,
                                                   (short)0, c0, false, false);
        c1 = __builtin_amdgcn_wmma_f32_16x16x4_f32(false, a, false, b1,
                                                   (short)0, c1, false, false);
        c2 = __builtin_amdgcn_wmma_f32_16x16x4_f32(false, a, false, b2,
                                                   (short)0, c2, false, false);
        c3 = __builtin_amdgcn_wmma_f32_16x16x4_f32(false, a, false, b3,
                                                   (short)0, c3, false, false);
    }

    const int mrow = m0 + ((lane >> 4) << 3);   // M = r + 8*(lane/16)
    const float bias0 = bias[n0 +  0 + row];
    const float bias1 = bias[n0 + 16 + row];
    const float bias2 = bias[n0 + 32 + row];
    const float bias3 = bias[n0 + 48 + row];

#pragma unroll
    for (int r = 0; r < 8; ++r) {
        const size_t base = (size_t)(mrow + r) * DD + row;
        O[base + n0 +  0] = c0[r] + bias0;
        O[base + n0 + 16] = c1[r] + bias1;
        O[base + n0 + 32] = c2[r] + bias2;
        O[base + n0 + 48] = c3[r] + bias3;
    }
}

// ---------------------------------------------------------------------------
// Kernel 2: fused scalar-energy softmax attention. One block per sample b.
// LDS stages k[b,:], v[b,:]; kmin/kmax give the exact softmax row max.
// ---------------------------------------------------------------------------
__global__ __launch_bounds__(256)
void attn_fused(const float* __restrict__ Q, const float* __restrict__ K,
                const float* __restrict__ V, const float* __restrict__ X,
                const float* __restrict__ gptr, float* __restrict__ out)
{
    __shared__ float kls[DD];
    __shared__ float vls[DD];
    __shared__ float rmax[256];
    __shared__ float rmin[256];

    const int b   = blockIdx.x;
    const int tid = threadIdx.x;

    float lmax = -3.4e38f, lmin = 3.4e38f;
    for (int i = tid; i < DD; i += 256) {
        const float kv = K[(size_t)b * DD + i];
        kls[i] = kv;
        vls[i] = V[(size_t)b * DD + i];
        lmax = fmaxf(lmax, kv);
        lmin = fminf(lmin, kv);
    }
    rmax[tid] = lmax;
    rmin[tid] = lmin;
    __syncthreads();
    for (int s = 128; s > 0; s >>= 1) {
        if (tid < s) {
            rmax[tid] = fmaxf(rmax[tid], rmax[tid + s]);
            rmin[tid] = fminf(rmin[tid], rmin[tid + s]);
        }
        __syncthreads();
    }
    const float kmax = rmax[0];
    const float kmin = rmin[0];
    const float gamma = gptr[0];

    const int wave = tid >> 5;   // 8 waves (wave32)
    const int lane = tid & 31;

    for (int j = wave; j < DD; j += 8) {
        const float s = Q[(size_t)b * DD + j];
        const float m = fmaxf(s * kmax, s * kmin);   // exact row max
        float se = 0.f, sev = 0.f;
        for (int i = lane; i < DD; i += 32) {
            const float e = __expf(fmaf(s, kls[i], -m));
            se  += e;
            sev += e * vls[i];
        }
#pragma unroll
        for (int off = 16; off > 0; off >>= 1) {
            se  += __shfl_xor(se,  off);
            sev += __shfl_xor(sev, off);
        }
        if (lane == 0) {
            out[(size_t)b * DD + j] =
                fmaf(gamma, sev / se, X[(size_t)b * DD + j]);
        }
    }
}

// ---------------------------------------------------------------------------
extern "C" void kernel_launch(void* const* d_in, const int* in_sizes, int n_in,
                              void* d_out, int out_size, void* d_ws,
                              size_t ws_size, hipStream_t stream)
{
    (void)in_sizes; (void)n_in; (void)out_size; (void)ws_size;
    const float* x     = (const float*)d_in[0];
    const float* Wq    = (const float*)d_in[1];
    const float* bq    = (const float*)d_in[2];
    const float* Wk    = (const float*)d_in[3];
    const float* bk    = (const float*)d_in[4];
    const float* Wv    = (const float*)d_in[5];
    const float* bv    = (const float*)d_in[6];
    const float* gamma = (const float*)d_in[7];
    float* out = (float*)d_out;

    float* Qs = (float*)d_ws;                       // 256*1024 floats
    float* Ks = Qs + (size_t)BB * DD;
    float* Vs = Ks + (size_t)BB * DD;

    dim3 grid1(BB / 16, DD / 64, 3);
    qkv_wmma_f32<<<grid1, 32, 0, stream>>>(x, Wq, bq, Wk, bk, Wv, bv,
                                           Qs, Ks, Vs);

    attn_fused<<<BB, 256, 0, stream>>>(Qs, Ks, Vs, x, gamma, out);
}